// GraphConvLayer_22840636080817
// MI455X (gfx1250) — compile-verified
//
#include <hip/hip_runtime.h>

typedef __attribute__((ext_vector_type(2))) float v2f;
typedef __attribute__((ext_vector_type(8))) float v8f;

#define CH 128        // IN_C == OUT_C == 128
#define XS_PITCH 132  // LDS pitch: 132 % 64 banks = 4 -> conflict-free column reads

// ---------------------------------------------------------------------------
// Pack W[128,128] into K-pair-major float2 layout:
//   Wp[(k>>1)*128 + n] = { W[k][n], W[k+1][n] }   (k even)
// so each WMMA B-operand fetch is one global_load_b64.
// ---------------------------------------------------------------------------
__global__ void pack_W(const float* __restrict__ W, float2* __restrict__ Wp) {
  const int i = blockIdx.x * blockDim.x + threadIdx.x;  // 0 .. 64*128-1
  if (i >= 64 * CH) return;
  const int k2 = i >> 7;          // K pair index
  const int n  = i & (CH - 1);
  float2 p;
  p.x = W[(size_t)(2 * k2) * CH + n];
  p.y = W[(size_t)(2 * k2 + 1) * CH + n];
  Wp[i] = p;
}

// ---------------------------------------------------------------------------
// GEMM: h[N,128] = x[N,128] @ W[128,128] using V_WMMA_F32_16X16X4_F32.
// One block = 8 waves = one 16-row strip; wave w computes the 16x16 tile at
// column 16w. Two independent accumulator chains over K.
// ---------------------------------------------------------------------------
__global__ __launch_bounds__(256) void gcn_gemm_wmma(
    const float* __restrict__ x, const float2* __restrict__ Wp,
    float* __restrict__ h, int n_nodes)
{
  __shared__ float Xs[16 * XS_PITCH];

  const int row0 = blockIdx.x * 16;

  // Stage the 16x128 strip of x into LDS (padded pitch).
  for (int i = threadIdx.x; i < 16 * CH; i += 256) {
    const int r = i >> 7;
    const int c = i & (CH - 1);
    const int gr = row0 + r;
    Xs[r * XS_PITCH + c] = (gr < n_nodes) ? x[(size_t)gr * CH + c] : 0.0f;
  }
  __syncthreads();

  const int wave = threadIdx.x >> 5;  // 0..7 -> column tile
  const int lane = threadIdx.x & 31;
  const int m    = lane & 15;         // A-matrix row within tile
  const int half = lane >> 4;         // 0 or 1
  const int kh   = half << 1;         // K sub-offset {0,2}
  const int ncol = wave * 16 + (lane & 15);  // global output column

  v8f acc0 = {};
  v8f acc1 = {};
  #pragma unroll 4
  for (int k = 0; k < CH; k += 8) {
    // --- K block [k, k+4) -> acc0
    {
      const int bk = k + kh;                 // even
      v2f a;
      a.x = Xs[m * XS_PITCH + bk];
      a.y = Xs[m * XS_PITCH + bk + 1];
      const float2 bp = Wp[(size_t)(bk >> 1) * CH + ncol];
      v2f b; b.x = bp.x; b.y = bp.y;
      acc0 = __builtin_amdgcn_wmma_f32_16x16x4_f32(
          false, a, false, b, (short)0, acc0, false, false);
    }
    // --- K block [k+4, k+8) -> acc1
    {
      const int bk = k + 4 + kh;             // even
      v2f a;
      a.x = Xs[m * XS_PITCH + bk];
      a.y = Xs[m * XS_PITCH + bk + 1];
      const float2 bp = Wp[(size_t)(bk >> 1) * CH + ncol];
      v2f b; b.x = bp.x; b.y = bp.y;
      acc1 = __builtin_amdgcn_wmma_f32_16x16x4_f32(
          false, a, false, b, (short)0, acc1, false, false);
    }
  }
  const v8f acc = acc0 + acc1;

  // D layout: VGPR i holds M = i + 8*half, N = lane&15.
  if (row0 + 16 <= n_nodes) {
    // Uniform fast path: straight-line clause of stores, no per-element exec
    // manipulation (always taken when N % 16 == 0).
    float* dst = h + (size_t)(row0 + half * 8) * CH + ncol;
    #pragma unroll
    for (int i = 0; i < 8; ++i) dst[(size_t)i * CH] = acc[i];
  } else {
    #pragma unroll
    for (int i = 0; i < 8; ++i) {
      const int gm = row0 + i + half * 8;
      if (gm < n_nodes) h[(size_t)gm * CH + ncol] = acc[i];
    }
  }
}

// ---------------------------------------------------------------------------
// Degree / normalization kernels
// ---------------------------------------------------------------------------
__global__ void init_deg(float* __restrict__ deg, int n) {
  const int i = blockIdx.x * blockDim.x + threadIdx.x;
  if (i < n) deg[i] = 1.0f;  // self-loop
}

__global__ void edge_deg(const int* __restrict__ col, float* __restrict__ deg,
                         int n_edges) {
  const int e = blockIdx.x * blockDim.x + threadIdx.x;
  if (e < n_edges) atomicAdd(&deg[col[e]], 1.0f);
}

__global__ void deg_to_dinv(float* __restrict__ deg, int n) {
  const int i = blockIdx.x * blockDim.x + threadIdx.x;
  if (i < n) deg[i] = rsqrtf(deg[i]);
}

// ---------------------------------------------------------------------------
// agg init: self-loop contribution + bias.  agg lives in d_out.
// ---------------------------------------------------------------------------
__global__ void init_agg(const float* __restrict__ h,
                         const float* __restrict__ dinv,
                         const float* __restrict__ bias,
                         float* __restrict__ agg, int n_nodes) {
  const size_t i = (size_t)blockIdx.x * blockDim.x + threadIdx.x;
  const size_t total = (size_t)n_nodes * CH;
  if (i >= total) return;
  const int r = (int)(i >> 7);
  const int c = (int)(i & (CH - 1));
  const float di = dinv[r];
  agg[i] = h[i] * di * di + bias[c];
}

// ---------------------------------------------------------------------------
// Edge scatter: one wave per edge, 4 channels per lane (float4 gather + atomics).
// ---------------------------------------------------------------------------
__global__ __launch_bounds__(256) void edge_scatter(
    const float* __restrict__ h, const float* __restrict__ dinv,
    const int* __restrict__ row_idx, const int* __restrict__ col_idx,
    float* __restrict__ agg, int n_edges)
{
  const int e = (int)(((size_t)blockIdx.x * blockDim.x + threadIdx.x) >> 5);
  const int lane = threadIdx.x & 31;
  if (e >= n_edges) return;
  const int r = row_idx[e];
  const int c = col_idx[e];
  const float norm = dinv[r] * dinv[c];
  const float4 v = ((const float4*)(h + (size_t)r * CH))[lane];
  float* dst = agg + (size_t)c * CH + lane * 4;
  atomicAdd(dst + 0, v.x * norm);
  atomicAdd(dst + 1, v.y * norm);
  atomicAdd(dst + 2, v.z * norm);
  atomicAdd(dst + 3, v.w * norm);
}

// ---------------------------------------------------------------------------
// BatchNorm partial sums: sums[0..127]=sum, sums[128..255]=sumsq.
// Channel-major thread mapping -> coalesced reads.
// ---------------------------------------------------------------------------
__global__ void zero_sums(float* __restrict__ sums) {
  if (threadIdx.x < 256) sums[threadIdx.x] = 0.0f;
}

__global__ __launch_bounds__(256) void bn_partial(
    const float* __restrict__ agg, float* __restrict__ sums, int n_nodes)
{
  const int c = threadIdx.x & (CH - 1);
  const int rsub = threadIdx.x >> 7;  // 0 or 1
  float s = 0.0f, s2 = 0.0f;
  for (int r = blockIdx.x * 2 + rsub; r < n_nodes; r += gridDim.x * 2) {
    const float v = agg[(size_t)r * CH + c];
    s += v;
    s2 += v * v;
  }
  __shared__ float red[256];
  red[threadIdx.x] = s;
  __syncthreads();
  float sTot = 0.0f;
  if (threadIdx.x < CH) sTot = red[threadIdx.x] + red[threadIdx.x + CH];
  __syncthreads();
  red[threadIdx.x] = s2;
  __syncthreads();
  if (threadIdx.x < CH) {
    const float s2Tot = red[threadIdx.x] + red[threadIdx.x + CH];
    atomicAdd(&sums[c], sTot);
    atomicAdd(&sums[CH + c], s2Tot);
  }
}

// ---------------------------------------------------------------------------
// Finalize: in-place BN (batch stats) + ReLU on d_out.
// ---------------------------------------------------------------------------
__global__ void bn_finalize(float* __restrict__ agg,
                            const float* __restrict__ sums,
                            const float* __restrict__ gamma,
                            const float* __restrict__ beta, int n_nodes) {
  const size_t i = (size_t)blockIdx.x * blockDim.x + threadIdx.x;
  const size_t total = (size_t)n_nodes * CH;
  if (i >= total) return;
  const int c = (int)(i & (CH - 1));
  const float invN = 1.0f / (float)n_nodes;
  const float mean = sums[c] * invN;
  const float var = sums[CH + c] * invN - mean * mean;
  const float y = (agg[i] - mean) * rsqrtf(var + 1e-5f) * gamma[c] + beta[c];
  agg[i] = fmaxf(y, 0.0f);
}

// ---------------------------------------------------------------------------
extern "C" void kernel_launch(void* const* d_in, const int* in_sizes, int n_in,
                              void* d_out, int out_size, void* d_ws, size_t ws_size,
                              hipStream_t stream) {
  const float* x     = (const float*)d_in[0];
  const float* W     = (const float*)d_in[1];
  const float* bias  = (const float*)d_in[2];
  const float* gamma = (const float*)d_in[3];
  const float* beta  = (const float*)d_in[4];
  const int* edge_index = (const int*)d_in[5];

  const int n_nodes = in_sizes[0] / CH;       // 50000
  const int n_edges = in_sizes[5] / 2;        // 800000
  const int* e_row = edge_index;              // edge_index[0,:]  (source)
  const int* e_col = edge_index + n_edges;    // edge_index[1,:]  (target)

  float* agg = (float*)d_out;                 // accumulate directly into output

  // Workspace layout: h [N*128] | dinv [N] | sums [256] | Wp [64*128 float2]
  char* ws = (char*)d_ws;
  float*  h    = (float*)ws;
  float*  dinv = (float*)(ws + (size_t)n_nodes * CH * sizeof(float));
  float*  sums = dinv + n_nodes;
  float2* Wp   = (float2*)(sums + 256);

  const size_t total = (size_t)n_nodes * CH;
  const int thr = 256;

  // 0) pack W into K-pair-major float2 layout
  pack_W<<<(64 * CH + thr - 1) / thr, thr, 0, stream>>>(W, Wp);

  // 1) degrees (with self-loop) -> dinv
  init_deg<<<(n_nodes + thr - 1) / thr, thr, 0, stream>>>(dinv, n_nodes);
  edge_deg<<<(n_edges + thr - 1) / thr, thr, 0, stream>>>(e_col, dinv, n_edges);
  deg_to_dinv<<<(n_nodes + thr - 1) / thr, thr, 0, stream>>>(dinv, n_nodes);

  // 2) h = x @ W  (WMMA f32)
  gcn_gemm_wmma<<<(n_nodes + 15) / 16, thr, 0, stream>>>(x, Wp, h, n_nodes);

  // 3) agg = self-loop + bias, then edge scatter-add
  init_agg<<<(int)((total + thr - 1) / thr), thr, 0, stream>>>(h, dinv, bias, agg, n_nodes);
  {
    const size_t threads_needed = (size_t)n_edges * 32;
    edge_scatter<<<(int)((threads_needed + thr - 1) / thr), thr, 0, stream>>>(
        h, dinv, e_row, e_col, agg, n_edges);
  }

  // 4) BatchNorm batch statistics
  zero_sums<<<1, thr, 0, stream>>>(sums);
  bn_partial<<<512, thr, 0, stream>>>(agg, sums, n_nodes);

  // 5) normalize + ReLU in place
  bn_finalize<<<(int)((total + thr - 1) / thr), thr, 0, stream>>>(agg, sums, gamma, beta, n_nodes);
}